// MSELoss_24386824307099
// MI455X (gfx1250) — compile-verified
//
#include <hip/hip_runtime.h>
#include <stdint.h>

#define B_    4
#define C_    16
#define NF_   10000
#define L_    100000
#define BC_   (B_ * C_)
#define BLK   256
#define BCCH  16                  // bc-slices per block (gridDim.y = BC_/BCCH = 4)
#define NPAIR (L_ / 2)            // 50000 l-pairs
#define NBX   ((NPAIR + BLK - 1) / BLK)   // 196
#define NBY   (BC_ / BCCH)                // 4

typedef __attribute__((ext_vector_type(2))) float v2f;
typedef __attribute__((ext_vector_type(8))) float v8f;

struct f2 { float x, y; };
struct __align__(16) f4 { float x, y, z, w; };
struct __align__(16) i4 { int   x, y, z, w; };

__global__ __launch_bounds__(BLK) void mse_main(
    const float* __restrict__ i_f, const float* __restrict__ i_s,
    const float* __restrict__ t_f, const float* __restrict__ t_s,
    const int*   __restrict__ xi,  const float* __restrict__ ks,
    float* __restrict__ partials)
{
  // 4-stage ring: [stage][0=i_s,1=t_s][tid*4 + {0..3}] ; 32 KB total
  __shared__ __align__(16) float smem[4][2][BLK * 4];
  __shared__ float wsum[BLK / 32];

  const int tid = threadIdx.x;
  const int p   = blockIdx.x * BLK + tid;            // l-pair index
  const int pc  = (p < NPAIR) ? p : (NPAIR - 1);     // clamp: keep EXEC all-1s
  const int l0  = pc * 2;                            // sites l0, l0+1

  const int bcb = blockIdx.y * BCCH;
  const int bce = bcb + BCCH;

  // One b128 each for indices / masks of both sites
  const i4 iv = *(const i4*)&xi[4 * pc];
  const f4 kv = *(const f4*)&ks[4 * pc];
  const bool fullA = (kv.x <= 0.0f) && (kv.y <= 0.0f);
  const bool fullB = (kv.z <= 0.0f) && (kv.w <= 0.0f);

  uint32_t ldsI[4], ldsT[4];
  #pragma unroll
  for (int s = 0; s < 4; ++s) {
    ldsI[s] = (uint32_t)(uintptr_t)&smem[s][0][tid * 4];
    ldsT[s] = (uint32_t)(uintptr_t)&smem[s][1][tid * 4];
  }

  // Prologue: stage bc = bcb .. bcb+2 into ring slots 0..2 (16B per lane)
  #pragma unroll
  for (int s = 0; s < 3; ++s) {
    const float* ga = i_s + ((size_t)(bcb + s) * L_ + l0) * 2;
    const float* gb = t_s + ((size_t)(bcb + s) * L_ + l0) * 2;
    asm volatile("global_load_async_to_lds_b128 %0, %1, off" :: "v"(ldsI[s]), "v"(ga) : "memory");
    asm volatile("global_load_async_to_lds_b128 %0, %1, off" :: "v"(ldsT[s]), "v"(gb) : "memory");
  }

  float acc = 0.0f;

  #pragma unroll 4
  for (int bc = bcb; bc < bce; ++bc) {
    const int i   = bc - bcb;
    const int buf = i & 3;

    if (bc + 3 < bce) {
      // ds_loads that last read ring slot (i+3)&3 happened at iteration i-1; let them land
      asm volatile("s_wait_dscnt 0x0" ::: "memory");
      const float* ga = i_s + ((size_t)(bc + 3) * L_ + l0) * 2;
      const float* gb = t_s + ((size_t)(bc + 3) * L_ + l0) * 2;
      asm volatile("global_load_async_to_lds_b128 %0, %1, off" :: "v"(ldsI[(i + 3) & 3]), "v"(ga) : "memory");
      asm volatile("global_load_async_to_lds_b128 %0, %1, off" :: "v"(ldsT[(i + 3) & 3]), "v"(gb) : "memory");
      // 8 outstanding; <=6 guarantees slot `buf`'s pair completed (in-order completion)
      asm volatile("s_wait_asynccnt 0x6" ::: "memory");
    } else {
      const int rem = bce - 1 - bc;                  // 2, 1, 0
      if      (rem == 2) asm volatile("s_wait_asynccnt 0x4" ::: "memory");
      else if (rem == 1) asm volatile("s_wait_asynccnt 0x2" ::: "memory");
      else               asm volatile("s_wait_asynccnt 0x0" ::: "memory");
    }

    const f4 sv = *(const f4*)&smem[buf][0][tid * 4]; // i_s: re0,im0,re1,im1
    const f4 tv = *(const f4*)&smem[buf][1][tid * 4]; // t_s

    // ---- site A (l0) ----
    {
      float gr = sv.x - tv.x;
      float gi = sv.y - tv.y;
      if (fullA) {
        const int g0 = (bc * NF_ + iv.x) * 2;
        const int g1 = (bc * NF_ + iv.y) * 2;
        f2 a0 = *(const f2*)&i_f[g0];
        f2 b0 = *(const f2*)&t_f[g0];
        f2 a1 = *(const f2*)&i_f[g1];
        f2 b1 = *(const f2*)&t_f[g1];
        if (bc + 1 < bce) {
          __builtin_prefetch(&i_f[g0 + NF_ * 2], 0, 1);
          __builtin_prefetch(&t_f[g0 + NF_ * 2], 0, 1);
          __builtin_prefetch(&i_f[g1 + NF_ * 2], 0, 1);
          __builtin_prefetch(&t_f[g1 + NF_ * 2], 0, 1);
        }
        const float f0r = a0.x - b0.x, f0i = a0.y - b0.y;
        const float f1r = a1.x - b1.x, f1i = a1.y - b1.y;
        gr -= (f0r * b1.x + f0i * b1.y) + (b0.x * f1r + b0.y * f1i);
        gi -= (f0i * b1.x - f0r * b1.y) + (b0.y * f1r - b0.x * f1i);
      }
      acc = fmaf(gr, gr, fmaf(gi, gi, acc));
    }
    // ---- site B (l0+1) ----
    {
      float gr = sv.z - tv.z;
      float gi = sv.w - tv.w;
      if (fullB) {
        const int g0 = (bc * NF_ + iv.z) * 2;
        const int g1 = (bc * NF_ + iv.w) * 2;
        f2 a0 = *(const f2*)&i_f[g0];
        f2 b0 = *(const f2*)&t_f[g0];
        f2 a1 = *(const f2*)&i_f[g1];
        f2 b1 = *(const f2*)&t_f[g1];
        if (bc + 1 < bce) {
          __builtin_prefetch(&i_f[g0 + NF_ * 2], 0, 1);
          __builtin_prefetch(&t_f[g0 + NF_ * 2], 0, 1);
          __builtin_prefetch(&i_f[g1 + NF_ * 2], 0, 1);
          __builtin_prefetch(&t_f[g1 + NF_ * 2], 0, 1);
        }
        const float f0r = a0.x - b0.x, f0i = a0.y - b0.y;
        const float f1r = a1.x - b1.x, f1i = a1.y - b1.y;
        gr -= (f0r * b1.x + f0i * b1.y) + (b0.x * f1r + b0.y * f1i);
        gi -= (f0i * b1.x - f0r * b1.y) + (b0.y * f1r - b0.x * f1i);
      }
      acc = fmaf(gr, gr, fmaf(gi, gi, acc));
    }
  }

  if (p >= NPAIR) acc = 0.0f;  // discard clamped lanes

  // ---- Wave reduction via V_WMMA_F32_16X16X4_F32 ----
  // A (16x4): VGPR0 = per-lane value, VGPR1 = 0. B = ones => D[m][n] = val[m] + val[m+16].
  // Sum of 8 D VGPRs gives half-sums in lanes 0-15 / 16-31; shfl_xor(16) completes it.
  v2f av; av[0] = acc;  av[1] = 0.0f;
  v2f bv; bv[0] = 1.0f; bv[1] = 1.0f;
  v8f cv = {0.0f, 0.0f, 0.0f, 0.0f, 0.0f, 0.0f, 0.0f, 0.0f};
  v8f dv = __builtin_amdgcn_wmma_f32_16x16x4_f32(false, av, false, bv, (short)0, cv, false, false);
  float s = dv[0] + dv[1] + dv[2] + dv[3] + dv[4] + dv[5] + dv[6] + dv[7];
  s += __shfl_xor(s, 16, 32);

  const int lane = tid & 31;
  const int wave = tid >> 5;
  if (lane == 0) wsum[wave] = s;
  __syncthreads();
  if (tid == 0) {
    float t = 0.0f;
    #pragma unroll
    for (int w = 0; w < BLK / 32; ++w) t += wsum[w];
    partials[blockIdx.y * gridDim.x + blockIdx.x] = t;
  }
}

__global__ __launch_bounds__(BLK) void mse_finalize(
    const float* __restrict__ partials, int n, float* __restrict__ out)
{
  __shared__ double red[BLK];
  double s = 0.0;
  for (int i = threadIdx.x; i < n; i += BLK) s += (double)partials[i];
  red[threadIdx.x] = s;
  __syncthreads();
  for (int off = BLK / 2; off > 0; off >>= 1) {
    if (threadIdx.x < off) red[threadIdx.x] += red[threadIdx.x + off];
    __syncthreads();
  }
  if (threadIdx.x == 0)
    out[0] = (float)(red[0] * (1.0 / ((double)B_ * (double)C_ * (double)L_)));
}

extern "C" void kernel_launch(void* const* d_in, const int* in_sizes, int n_in,
                              void* d_out, int out_size, void* d_ws, size_t ws_size,
                              hipStream_t stream) {
  const float* i_f = (const float*)d_in[0];
  const float* i_s = (const float*)d_in[1];
  const float* t_f = (const float*)d_in[2];
  const float* t_s = (const float*)d_in[3];
  const int*   xi  = (const int*)  d_in[4];
  const float* ks  = (const float*)d_in[5];
  float* out      = (float*)d_out;
  float* partials = (float*)d_ws;

  dim3 grid(NBX, NBY);                         // 196 x 4 = 784 blocks
  mse_main<<<grid, BLK, 0, stream>>>(i_f, i_s, t_f, t_s, xi, ks, partials);
  mse_finalize<<<1, BLK, 0, stream>>>(partials, NBX * NBY, out);
}